// MultiresHashTableEncoding_79731772883612
// MI455X (gfx1250) — compile-verified
//
#include <hip/hip_runtime.h>
#include <cstdint>

#define NPTS   (1u << 20)
#define TMASK  ((1u << 19) - 1u)

typedef float v4f  __attribute__((ext_vector_type(4)));
typedef int   v4i  __attribute__((ext_vector_type(4)));
// 16-byte payloads that are only guaranteed 8-byte aligned (dense corner pairs)
typedef float v4f8 __attribute__((ext_vector_type(4), aligned(8)));

struct Tables { const float* t[16]; };

// p must already be clamped to [0,1]
__device__ __forceinline__ void axis_setup(float f, int g, uint32_t& i0, float& cf)
{
    float fi = (float)(g - 1) * f;
    float fl = fminf(floorf(fi), (float)(g - 2));
    cf = fi - fl;
    i0 = (uint32_t)fl;
}

// Dense level: table entries for corners (X=ix, X=ix+1) are contiguous -> one 16B load per (y,z) pair.
__device__ __forceinline__ void encode_dense(float fx, float fy, float fz, int g,
                                             const float* __restrict__ tbl,
                                             float& f0, float& f1)
{
    uint32_t ix, iy, iz; float cx, cy, cz;
    axis_setup(fx, g, ix, cx);
    axis_setup(fy, g, iy, cy);
    axis_setup(fz, g, iz, cz);

    uint32_t ug = (uint32_t)g;
    uint32_t ty[2] = { ug * iy,      ug * (iy + 1u) };
    uint32_t tz[2] = { ug * ug * iz, ug * ug * (iz + 1u) };
    float    wy[2] = { 1.0f - cy, cy };
    float    wz[2] = { 1.0f - cz, cz };
    float    wx0 = 1.0f - cx, wx1 = cx;

    f0 = 0.0f; f1 = 0.0f;
    #pragma unroll
    for (int oy = 0; oy < 2; ++oy)
    #pragma unroll
    for (int oz = 0; oz < 2; ++oz) {
        uint32_t b   = ix + ty[oy] + tz[oz];          // entry index of (ix, oy, oz)
        float    wyz = wy[oy] * wz[oz];
        v4f8 v = *(const v4f8*)(tbl + 2u * b);        // entries b and b+1 (16B)
        float w0 = wx0 * wyz, w1 = wx1 * wyz;
        f0 = fmaf(w0, v.x, fmaf(w1, v.z, f0));
        f1 = fmaf(w0, v.y, fmaf(w1, v.w, f1));
    }
}

// Hashed level: spatial hash, table size is 2^19 so % becomes mask.
__device__ __forceinline__ void encode_hashed(float fx, float fy, float fz, int g,
                                              const float* __restrict__ tbl,
                                              float& f0, float& f1)
{
    uint32_t ix, iy, iz; float cx, cy, cz;
    axis_setup(fx, g, ix, cx);
    axis_setup(fy, g, iy, cy);
    axis_setup(fz, g, iz, cz);

    uint32_t ty[2] = { 19349663u * iy, 19349663u * (iy + 1u) };
    uint32_t tz[2] = { 83492791u * iz, 83492791u * (iz + 1u) };
    float    wy[2] = { 1.0f - cy, cy };
    float    wz[2] = { 1.0f - cz, cz };
    float    wx[2] = { 1.0f - cx, cx };

    f0 = 0.0f; f1 = 0.0f;
    #pragma unroll
    for (int oy = 0; oy < 2; ++oy)
    #pragma unroll
    for (int oz = 0; oz < 2; ++oz) {
        uint32_t tyz = ty[oy] ^ tz[oz];               // factored hash term
        float    wyz = wy[oy] * wz[oz];
        #pragma unroll
        for (int ox = 0; ox < 2; ++ox) {
            uint32_t idx = ((ix + (uint32_t)ox) ^ tyz) & TMASK;
            float2 v = *(const float2*)(tbl + 2u * idx);   // 32-bit index -> saddr+scaled voffset
            float  w = wx[ox] * wyz;
            f0 = fmaf(w, v.x, f0);
            f1 = fmaf(w, v.y, f1);
        }
    }
}

__global__ void __launch_bounds__(256)
mrhash_kernel(const float* __restrict__ x, Tables tabs, float* __restrict__ out)
{
    // Level-0 dense table: 16^3 entries * 2 floats = 32 KB, staged in LDS.
    __shared__ __align__(16) float tbl0[16 * 16 * 16 * 2];

    {
        const float* src = tabs.t[0];
        int t = threadIdx.x;
#if __has_builtin(__builtin_amdgcn_global_load_async_to_lds_b128)
        typedef __attribute__((address_space(1))) v4i GInt4;   // global int4
        typedef __attribute__((address_space(3))) v4i LInt4;   // LDS int4
        #pragma unroll
        for (int k = 0; k < 8; ++k) {
            int fo = (t + 256 * k) * 4;   // float index, 16B granules
            __builtin_amdgcn_global_load_async_to_lds_b128(
                (GInt4*)(src + fo), (LInt4*)(tbl0 + fo), 0, 0);
        }
#if __has_builtin(__builtin_amdgcn_s_wait_asynccnt)
        __builtin_amdgcn_s_wait_asynccnt(0);
#else
        asm volatile("s_wait_asynccnt 0" ::: "memory");
#endif
#else
        #pragma unroll
        for (int k = 0; k < 8; ++k) {
            int fo = t + 256 * k;
            ((v4f*)tbl0)[fo] = ((const v4f*)src)[fo];
        }
#endif
        __syncthreads();
    }

    const uint32_t stride = gridDim.x * blockDim.x;
    for (uint32_t i = blockIdx.x * blockDim.x + threadIdx.x; i < NPTS; i += stride) {
        const float* xp = x + 3u * (size_t)i;
        float px = __builtin_nontemporal_load(xp + 0);
        float py = __builtin_nontemporal_load(xp + 1);
        float pz = __builtin_nontemporal_load(xp + 2);

        // Clamp once, reused by all 16 levels.
        float fx = fminf(fmaxf(px, 0.0f), 1.0f);
        float fy = fminf(fmaxf(py, 0.0f), 1.0f);
        float fz = fminf(fmaxf(pz, 0.0f), 1.0f);

        float o[32];
        float f0, f1;

        // Dense levels (g^3 <= 2^19): 16 (LDS), 23, 32, 45, 64
        encode_dense (fx, fy, fz,   16, tbl0,       f0, f1); o[ 0]=f0; o[ 1]=f1;
        encode_dense (fx, fy, fz,   23, tabs.t[ 1], f0, f1); o[ 2]=f0; o[ 3]=f1;
        encode_dense (fx, fy, fz,   32, tabs.t[ 2], f0, f1); o[ 4]=f0; o[ 5]=f1;
        encode_dense (fx, fy, fz,   45, tabs.t[ 3], f0, f1); o[ 6]=f0; o[ 7]=f1;
        encode_dense (fx, fy, fz,   64, tabs.t[ 4], f0, f1); o[ 8]=f0; o[ 9]=f1;
        // Hashed levels (g^3 > 2^19)
        encode_hashed(fx, fy, fz,   91, tabs.t[ 5], f0, f1); o[10]=f0; o[11]=f1;
        encode_hashed(fx, fy, fz,  128, tabs.t[ 6], f0, f1); o[12]=f0; o[13]=f1;
        encode_hashed(fx, fy, fz,  181, tabs.t[ 7], f0, f1); o[14]=f0; o[15]=f1;
        encode_hashed(fx, fy, fz,  256, tabs.t[ 8], f0, f1); o[16]=f0; o[17]=f1;
        encode_hashed(fx, fy, fz,  362, tabs.t[ 9], f0, f1); o[18]=f0; o[19]=f1;
        encode_hashed(fx, fy, fz,  512, tabs.t[10], f0, f1); o[20]=f0; o[21]=f1;
        encode_hashed(fx, fy, fz,  724, tabs.t[11], f0, f1); o[22]=f0; o[23]=f1;
        encode_hashed(fx, fy, fz, 1024, tabs.t[12], f0, f1); o[24]=f0; o[25]=f1;
        encode_hashed(fx, fy, fz, 1448, tabs.t[13], f0, f1); o[26]=f0; o[27]=f1;
        encode_hashed(fx, fy, fz, 2048, tabs.t[14], f0, f1); o[28]=f0; o[29]=f1;
        encode_hashed(fx, fy, fz, 2896, tabs.t[15], f0, f1); o[30]=f0; o[31]=f1;

        // Streaming output: non-temporal b128 stores so the 128MB output
        // stream does not evict the L2-resident hash tables.
        float* obase = out + (size_t)i * 32u;
        #pragma unroll
        for (int k = 0; k < 8; ++k) {
            v4f v = { o[4*k + 0], o[4*k + 1], o[4*k + 2], o[4*k + 3] };
            __builtin_nontemporal_store(v, (v4f*)obase + k);
        }
    }
}

extern "C" void kernel_launch(void* const* d_in, const int* in_sizes, int n_in,
                              void* d_out, int out_size, void* d_ws, size_t ws_size,
                              hipStream_t stream) {
    (void)in_sizes; (void)n_in; (void)d_ws; (void)ws_size; (void)out_size;
    const float* x = (const float*)d_in[0];
    Tables tabs;
    #pragma unroll
    for (int l = 0; l < 16; ++l) tabs.t[l] = (const float*)d_in[1 + l];
    float* out = (float*)d_out;

    dim3 grid(2048), block(256);   // 16K waves for gather latency hiding; 2 pts/thread
    hipLaunchKernelGGL(mrhash_kernel, grid, block, 0, stream, x, tabs, out);
}